// PLEk_64321430225120
// MI455X (gfx1250) — compile-verified
//
#include <hip/hip_runtime.h>
#include <math.h>

// Problem constants (from reference)
#define B_     8
#define CIN    256
#define COUT   256
#define KK     4
#define H_     128
#define W_     128
#define NPTS   512
#define NP_TOTAL (B_ * NPTS)   // 4096 output points

typedef float v2f __attribute__((ext_vector_type(2)));
typedef float v8f __attribute__((ext_vector_type(8)));

#define LDS_STRIDE 260  // 256 + 4 pad -> bank-conflict-free column reads

__global__ __launch_bounds__(256) void ple_fused_kernel(
    const float* __restrict__ input,   // [B, CIN, H, W]
    const float* __restrict__ ax,      // [B, NPTS]
    const float* __restrict__ ay,      // [B, NPTS]
    const float* __restrict__ Wp,      // [COUT, CIN]
    const float* __restrict__ bp,      // [COUT]
    const float* __restrict__ Woff,    // [K, 2, CIN]
    const float* __restrict__ boff,    // [K, 2]
    const float* __restrict__ Ww,      // [K, CIN]
    const float* __restrict__ bw,      // [K]
    float* __restrict__ out)           // [B, NPTS, COUT]
{
    __shared__ float sX[16 * LDS_STRIDE];  // xeff for 16 points
    __shared__ float sWsum[16];            // total valid weight per point

    const int tid   = threadIdx.x;
    const int wave  = tid >> 5;    // 0..7
    const int lane  = tid & 31;
    const int mBase = blockIdx.x * 16;  // first point of this tile

    // ---------------- Phase 1: build effective sampled vectors ----------------
    for (int ptSub = 0; ptSub < 2; ++ptSub) {
        const int pt      = wave * 2 + ptSub;   // 0..15 in tile
        const int pGlobal = mBase + pt;         // 0..4095
        const int b = pGlobal / NPTS;
        const int p = pGlobal - b * NPTS;

        const float axv = ax[b * NPTS + p];
        const float ayv = ay[b * NPTS + p];
        const int x = (int)axv;   // trunc (coords are >= 0)
        const int y = (int)ayv;

        const float* inB = input + (size_t)b * CIN * H_ * W_;

        // this lane owns channels c = lane + 32*i
        float xv[8];
        #pragma unroll
        for (int i = 0; i < 8; ++i) {
            const int c = lane + 32 * i;
            xv[i] = inB[((size_t)c * H_ + y) * W_ + x];
        }

        // 12 dot products: j=0..7 -> Woff[k][d] (j=k*2+d); j=8..11 -> Ww[k]
        float part[12];
        #pragma unroll
        for (int j = 0; j < 8; ++j) {
            const float* wrow = Woff + j * CIN;
            float s = 0.f;
            #pragma unroll
            for (int i = 0; i < 8; ++i) s += xv[i] * wrow[lane + 32 * i];
            part[j] = s;
        }
        #pragma unroll
        for (int j = 0; j < 4; ++j) {
            const float* wrow = Ww + j * CIN;
            float s = 0.f;
            #pragma unroll
            for (int i = 0; i < 8; ++i) s += xv[i] * wrow[lane + 32 * i];
            part[8 + j] = s;
        }
        // butterfly reduce across the wave (wave32)
        #pragma unroll
        for (int off = 16; off >= 1; off >>= 1) {
            #pragma unroll
            for (int j = 0; j < 12; ++j)
                part[j] += __shfl_xor(part[j], off, 32);
        }

        float fx[KK], fy[KK], lg[KK];
        #pragma unroll
        for (int k = 0; k < KK; ++k) {
            fx[k] = part[k * 2 + 0] + boff[k * 2 + 0];
            fy[k] = part[k * 2 + 1] + boff[k * 2 + 1];
            lg[k] = part[8 + k] + bw[k];
        }
        // softmax over K
        float mx = fmaxf(fmaxf(lg[0], lg[1]), fmaxf(lg[2], lg[3]));
        float ek[KK]; float se = 0.f;
        #pragma unroll
        for (int k = 0; k < KK; ++k) { ek[k] = __expf(lg[k] - mx); se += ek[k]; }
        const float inv = 1.f / se;

        // weighted corner gathers -> xeff
        float acc[8];
        #pragma unroll
        for (int i = 0; i < 8; ++i) acc[i] = 0.f;
        float wsum = 0.f;
        const float sxs = (float)W_ / (float)(W_ - 1);
        const float sys = (float)H_ / (float)(H_ - 1);

        #pragma unroll
        for (int k = 0; k < KK; ++k) {
            const float wk = ek[k] * inv;
            const float sx = ((float)x + fx[k]) * sxs - 0.5f;
            const float sy = ((float)y + fy[k]) * sys - 0.5f;
            const float x0f = floorf(sx), y0f = floorf(sy);
            const int x0 = (int)x0f, y0 = (int)y0f;
            const float wx1 = sx - x0f, wy1 = sy - y0f;
            const float cw[4] = { (1.f - wx1) * (1.f - wy1), wx1 * (1.f - wy1),
                                  (1.f - wx1) * wy1,         wx1 * wy1 };
            const int cx[4] = { x0, x0 + 1, x0,     x0 + 1 };
            const int cy[4] = { y0, y0,     y0 + 1, y0 + 1 };
            #pragma unroll
            for (int c = 0; c < 4; ++c) {
                if (cx[c] >= 0 && cx[c] < W_ && cy[c] >= 0 && cy[c] < H_) {
                    const float w = wk * cw[c];
                    const float* pix = inB + (size_t)cy[c] * W_ + cx[c];
                    #pragma unroll
                    for (int i = 0; i < 8; ++i)
                        acc[i] += w * pix[(size_t)(lane + 32 * i) * (H_ * W_)];
                    wsum += w;
                }
            }
        }

        #pragma unroll
        for (int i = 0; i < 8; ++i)
            sX[pt * LDS_STRIDE + lane + 32 * i] = acc[i];
        if (lane == 0) sWsum[pt] = wsum;
    }

    __syncthreads();

    // ---------------- Phase 2: Out[16, 256] = Xeff(LDS) @ Wp^T + bp*wsum ------
    // V_WMMA_F32_16X16X4_F32 fragments:
    //   A 16x4 : lane = 16*half + m ; v0 = A[m][2*half], v1 = A[m][2*half+1]
    //   B 4x16 : lane = 16*half + n ; v0 = B[2*half][n], v1 = B[2*half+1][n]
    //   C 16x16: VGPR r -> row = r + 8*half, col = lane&15
    const int half = lane >> 4;
    const int mrow = lane & 15;

    for (int t = 0; t < 2; ++t) {
        const int nBase = wave * 16 + t * 128;   // two N-tiles per wave
        const int col   = nBase + mrow;

        const float bias = bp[col];
        v8f c;
        #pragma unroll
        for (int r = 0; r < 8; ++r)
            c[r] = bias * sWsum[r + 8 * half];

        for (int kb = 0; kb < CIN; kb += 4) {
            const int kk = kb + 2 * half;
            v2f a, bf;
            a.x = sX[mrow * LDS_STRIDE + kk];
            a.y = sX[mrow * LDS_STRIDE + kk + 1];
            const float* wpp = Wp + (size_t)col * CIN + kk;
            bf.x = wpp[0];
            bf.y = wpp[1];
            c = __builtin_amdgcn_wmma_f32_16x16x4_f32(
                    false, a, false, bf, (short)0, c, false, false);
        }

        #pragma unroll
        for (int r = 0; r < 8; ++r) {
            const int row = mBase + r + 8 * half;
            out[(size_t)row * COUT + col] = c[r];
        }
    }
}

extern "C" void kernel_launch(void* const* d_in, const int* in_sizes, int n_in,
                              void* d_out, int out_size, void* d_ws, size_t ws_size,
                              hipStream_t stream) {
    const float* input = (const float*)d_in[0];
    const float* ax    = (const float*)d_in[1];
    const float* ay    = (const float*)d_in[2];
    const float* Wp    = (const float*)d_in[3];
    const float* bp    = (const float*)d_in[4];
    const float* Woff  = (const float*)d_in[5];
    const float* boff  = (const float*)d_in[6];
    const float* Ww    = (const float*)d_in[7];
    const float* bw    = (const float*)d_in[8];
    float* out = (float*)d_out;

    dim3 grid(NP_TOTAL / 16);   // 256 blocks, 16 points each
    dim3 block(256);            // 8 wave32 waves
    ple_fused_kernel<<<grid, block, 0, stream>>>(input, ax, ay, Wp, bp,
                                                 Woff, boff, Ww, bw, out);
}